// DiskKinematics_4741643894785
// MI455X (gfx1250) — compile-verified
//
#include <hip/hip_runtime.h>
#include <hip/hip_bf16.h>
#include <math.h>

// ---- problem constants (from reference) ----
#define NBINS   50
#define NFEAT   8                    // 7 used features + 1 pad slot
#define FLATN   (NBINS * NFEAT)      // 400 flat (bin,feat) entries
#define INV_DR  5.0f                 // 1/DR, DR = (10-0)/50 = 0.2

// ---- tuning ----
#define BLOCK       256              // 8 waves (wave32)
#define NCOPIES     8                // LDS histogram replication
#define COPY_STRIDE (FLATN + 1)      // 401: 401 mod 64 = 17 (coprime w/ 64 banks)
#define P_MAX       512              // max partial tables (multiple of 4)

typedef __attribute__((ext_vector_type(2))) float v2f;
typedef __attribute__((ext_vector_type(8))) float v8f;

#if __has_builtin(__builtin_amdgcn_wmma_f32_16x16x4_f32)
#define HAVE_WMMA_F32X4 1
#else
#define HAVE_WMMA_F32X4 0
#endif

// ---------------------------------------------------------------------------
// Kernel 1: bandwidth-bound streaming histogram.
// Grid-stride over particles; LDS-privatized (8 copies) f32 histograms via
// ds_add_f32 atomics; per-block partial table (400 floats) written to d_ws.
// ---------------------------------------------------------------------------
__global__ __launch_bounds__(BLOCK) void disk_hist_kernel(
    const float* __restrict__ pos, const float* __restrict__ vel,
    const float* __restrict__ mass, float* __restrict__ partials, int n)
{
    __shared__ float sh[NCOPIES * COPY_STRIDE];   // 3208 floats = 12.8 KB
    for (int i = threadIdx.x; i < NCOPIES * COPY_STRIDE; i += BLOCK)
        sh[i] = 0.0f;
    __syncthreads();

    float* my = sh + (threadIdx.x & (NCOPIES - 1)) * COPY_STRIDE;

    const int stride = gridDim.x * BLOCK;
    for (int i = blockIdx.x * BLOCK + threadIdx.x; i < n; i += stride) {
        const float x  = pos[3 * i + 0];
        const float y  = pos[3 * i + 1];
        const float vx = vel[3 * i + 0];
        const float vy = vel[3 * i + 1];
        const float vz = vel[3 * i + 2];
        const float w  = mass[i];

        const float r = sqrtf(fmaf(x, x, y * y));
        const int   b = (int)floorf(r * INV_DR);      // r >= 0 -> b >= 0
        if (b < NBINS) {
            const float inv_r = 1.0f / r;
            const float vr = fmaf(x, vx,  y * vy) * inv_r;
            const float vp = fmaf(y, vx, -x * vy) * inv_r;
            float* h = my + b * NFEAT;
            atomicAdd(h + 0, w);                       // ds_add_f32
            atomicAdd(h + 1, w * vr);
            atomicAdd(h + 2, w * vr * vr);
            atomicAdd(h + 3, w * vp);
            atomicAdd(h + 4, w * vp * vp);
            atomicAdd(h + 5, w * vz);
            atomicAdd(h + 6, w * vz * vz);
        }
    }
    __syncthreads();

    // fold the 8 LDS copies -> per-block partial table in scratch
    float* outp = partials + (size_t)blockIdx.x * FLATN;
    for (int idx = threadIdx.x; idx < FLATN; idx += BLOCK) {
        float s = 0.0f;
#pragma unroll
        for (int c = 0; c < NCOPIES; ++c) s += sh[c * COPY_STRIDE + idx];
        outp[idx] = s;
    }
}

// ---------------------------------------------------------------------------
// Kernel 2: reduce P partial tables with V_WMMA_F32_16X16X4_F32 and finalize.
// One wave per 16-column tile (25 blocks; tile t covers bins 2t, 2t+1).
// A = ones(16x4) f32  ->  D[m][n] = sum_k B[k][n]  (exact f32, layout-robust).
// B lane L supplies partial rows {p+rowsel, p+rowsel+1} of column (t*16+L&15).
// ---------------------------------------------------------------------------
__global__ __launch_bounds__(32) void disk_reduce_kernel(
    const float* __restrict__ partials, float* __restrict__ out, int P)
{
    const int lane   = threadIdx.x;
    const int tile   = blockIdx.x;            // 0..24
    const int col    = tile * 16 + (lane & 15);
    const int rowsel = (lane >> 4) * 2;       // lanes 0-15 -> rows k0,k1; 16-31 -> k2,k3
    __shared__ float tot[16];

#if HAVE_WMMA_F32X4
    v8f c = {0.f, 0.f, 0.f, 0.f, 0.f, 0.f, 0.f, 0.f};
    v2f a = {1.0f, 1.0f};                     // all-ones A matrix
    for (int p = 0; p < P; p += 4) {
        v2f b;
        b[0] = partials[(size_t)(p + rowsel    ) * FLATN + col];
        b[1] = partials[(size_t)(p + rowsel + 1) * FLATN + col];
        // v_wmma_f32_16x16x4_f32: D = A*B + C (f32 throughout)
        c = __builtin_amdgcn_wmma_f32_16x16x4_f32(
                /*neg_a=*/false, a, /*neg_b=*/false, b,
                /*c_mod=*/(short)0, c, /*reuse_a=*/false, /*reuse_b=*/false);
    }
    if (lane < 16) tot[lane] = c[0];          // D row 0 = column totals
#else
    if (lane < 16) {
        float s = 0.0f;
        for (int p = 0; p < P; ++p) s += partials[(size_t)p * FLATN + col];
        tot[lane] = s;
    }
#endif
    __syncthreads();

    // finalize: this tile owns bins 2*tile + {0,1}
    if (lane < 2) {
        const float* t   = tot + lane * NFEAT;
        const float inv  = 1.0f / t[0];
        const float vr_m = t[1] * inv, vr2 = t[2] * inv;
        const float vp_m = t[3] * inv, vp2 = t[4] * inv;
        const float vz_m = t[5] * inv, vz2 = t[6] * inv;
        const int   b    = tile * 2 + lane;
        out[0 * NBINS + b] = vp_m;
        out[1 * NBINS + b] = sqrtf(vp2 - vp_m * vp_m);
        out[2 * NBINS + b] = vr_m;
        out[3 * NBINS + b] = sqrtf(vr2 - vr_m * vr_m);
        out[4 * NBINS + b] = vz_m;
        out[5 * NBINS + b] = sqrtf(vz2 - vz_m * vz_m);
    }
}

// ---------------------------------------------------------------------------
extern "C" void kernel_launch(void* const* d_in, const int* in_sizes, int n_in,
                              void* d_out, int out_size, void* d_ws, size_t ws_size,
                              hipStream_t stream) {
    const float* pos  = (const float*)d_in[0];  // (N,3) f32
    const float* vel  = (const float*)d_in[1];  // (N,3) f32
    const float* mass = (const float*)d_in[2];  // (N,)  f32
    float*       out  = (float*)d_out;          // (6,50) f32
    const int    n    = in_sizes[2];            // N

    float* partials = (float*)d_ws;
    int P = (int)(ws_size / (FLATN * sizeof(float)));
    if (P > P_MAX) P = P_MAX;
    P &= ~3;                                    // K-loop steps of 4
    if (P < 4) P = 4;

    disk_hist_kernel<<<P, BLOCK, 0, stream>>>(pos, vel, mass, partials, n);
    disk_reduce_kernel<<<FLATN / 16, 32, 0, stream>>>(partials, out, P);
}